// LatentModulatedSIRENLayer_spatial_84524956385522
// MI455X (gfx1250) — compile-verified
//
#include <hip/hip_runtime.h>
#include <hip/hip_bf16.h>
#include <cstddef>

typedef __attribute__((ext_vector_type(2))) float v2f;
typedef __attribute__((ext_vector_type(8))) float v8f;

#define SIREN_W0 30.0f
#define BB 4
#define NN 65536
#define CC 64
#define HWSZ 1024         // 32*32
#define OO 512

// ---------------------------------------------------------------------------
// Kernel 1: modulation maps via f32 WMMA.
//   smap[b*HW + hw][o] = sum_c latent[b][c][hw] * W_shift[o][c] + b_shift[o]
//   vmap  likewise from v / W_vshift / b_vshift
// One wave per 16x16 output tile; K-loop of 16 x V_WMMA_F32_16X16X4_F32.
// ---------------------------------------------------------------------------
__global__ __launch_bounds__(128)
void modmap_wmma_kernel(const float* __restrict__ latent,   // [B, C, HW]
                        const float* __restrict__ vsrc,     // [B, C, HW]
                        const float* __restrict__ Wshift,   // [O, C]
                        const float* __restrict__ bshift,   // [O]
                        const float* __restrict__ Wvshift,  // [O, C]
                        const float* __restrict__ bvshift,  // [O]
                        float* __restrict__ smap,           // [B*HW, O]
                        float* __restrict__ vmap)           // [B*HW, O]
{
    const int wave  = threadIdx.x >> 5;          // 0..3
    const int lane  = threadIdx.x & 31;
    const int half  = lane >> 4;                 // 0 or 1 (K sub-block / M sub-block)
    const int l16   = lane & 15;

    const int ntile = blockIdx.x * 4 + wave;     // 0..31  (O / 16)
    const int mtile = blockIdx.y;                // 0..255 (B*HW / 16)
    const int map   = blockIdx.z;                // 0 = shift, 1 = shift_v

    const float* __restrict__ A    = map ? vsrc    : latent;
    const float* __restrict__ Wm   = map ? Wvshift : Wshift;
    const float* __restrict__ bias = map ? bvshift : bshift;
    float*       __restrict__ dst  = map ? vmap    : smap;

    // A operand: lane holds row m = l16 (both halves), K pair {k, k+1} with
    // lanes 16-31 shifted by +2 in K (per ISA 32-bit A 16x4 layout).
    const int mrow = mtile * 16 + l16;           // global row = b*HW + hw
    const int b    = mrow >> 10;                 // / 1024
    const int hw   = mrow & (HWSZ - 1);
    const float* __restrict__ Abase = A + ((size_t)b * CC) * HWSZ + hw;

    // B operand: lane holds col n = l16, same K split as A (mirror layout).
    const int ncol = ntile * 16 + l16;
    const float* __restrict__ Wrow = Wm + (size_t)ncol * CC;

    v8f acc = {};
    #pragma unroll
    for (int k = 0; k < CC; k += 4) {
        const int ka = k + half * 2;
        v2f a, bmat;
        a.x    = Abase[(size_t)ka * HWSZ];
        a.y    = Abase[(size_t)(ka + 1) * HWSZ];
        bmat.x = Wrow[ka];
        bmat.y = Wrow[ka + 1];
        acc = __builtin_amdgcn_wmma_f32_16x16x4_f32(
                /*neg_a=*/false, a, /*neg_b=*/false, bmat,
                /*c_mod=*/(short)0, acc, /*reuse_a=*/false, /*reuse_b=*/false);
    }

    // C/D layout: VGPR j -> row (mtile*16 + j) for lanes 0-15, (+8) for 16-31,
    // col = ntile*16 + l16.
    const int   row0 = mtile * 16 + half * 8;
    const int   col  = ntile * 16 + l16;
    const float bv   = bias[col];
    #pragma unroll
    for (int j = 0; j < 8; ++j) {
        dst[(size_t)(row0 + j) * OO + col] = acc[j] + bv;
    }
}

// ---------------------------------------------------------------------------
// Kernel 2: per-point gather + SIREN activation (store-bandwidth bound).
// 128 threads x 4 channels = 512 channels; 32 points per block.
// ---------------------------------------------------------------------------
__global__ __launch_bounds__(128)
void siren_gather_kernel(const float* __restrict__ x,      // [B*N, 2]
                         const float* __restrict__ Wlin,   // [O, 2]
                         const float* __restrict__ blin,   // [O]
                         const float* __restrict__ smap,   // [B*HW, O]
                         const float* __restrict__ vmap,   // [B*HW, O]
                         float* __restrict__ out)          // [B*N, O]
{
    const int tid = threadIdx.x;                 // 0..127
    const int o0  = tid * 4;

    // Per-thread SIREN weights for its 4 channels (held in registers).
    const float4 wA = ((const float4*)Wlin)[tid * 2];      // rows o0, o0+1
    const float4 wB = ((const float4*)Wlin)[tid * 2 + 1];  // rows o0+2, o0+3
    const float4 bl = ((const float4*)blin)[tid];

    // Coordinate weights: rows 0 and 1 of W_lin (uniform).
    const float wl00 = Wlin[0], wl01 = Wlin[1];
    const float wl10 = Wlin[2], wl11 = Wlin[3];
    const float bl0  = blin[0], bl1  = blin[1];

    const int PPB   = 32;
    const int pbase = blockIdx.x * PPB;

    for (int i = 0; i < PPB; ++i) {
        const int p = pbase + i;                 // global point (uniform in block)
        const float x0 = x[(size_t)p * 2];
        const float x1 = x[(size_t)p * 2 + 1];

        // h channels 0/1 -> nearest-neighbor grid coords (round-half-even).
        const float h0 = fmaf(x1, wl01, fmaf(x0, wl00, bl0));
        const float h1 = fmaf(x1, wl11, fmaf(x0, wl10, bl1));
        const float fx = rintf(fmaf(h0, 15.5f, 15.5f));    // (h0+1)/2*31
        const float fy = rintf(fmaf(h1, 15.5f, 15.5f));
        int ix = (int)fx; ix = ix < 0 ? 0 : (ix > 31 ? 31 : ix);
        int iy = (int)fy; iy = iy < 0 ? 0 : (iy > 31 ? 31 : iy);

        const int    b    = p >> 16;             // p / N
        const size_t gidx = ((size_t)(b * HWSZ + iy * 32 + ix)) * OO + o0;

        const float4 s  = *(const float4*)(smap + gidx);
        const float4 sv = *(const float4*)(vmap + gidx);

        float4 r;
        r.x = __sinf(SIREN_W0 * (fmaf(x0, wA.x, fmaf(x1, wA.y, bl.x)) + s.x + sv.x));
        r.y = __sinf(SIREN_W0 * (fmaf(x0, wA.z, fmaf(x1, wA.w, bl.y)) + s.y + sv.y));
        r.z = __sinf(SIREN_W0 * (fmaf(x0, wB.x, fmaf(x1, wB.y, bl.z)) + s.z + sv.z));
        r.w = __sinf(SIREN_W0 * (fmaf(x0, wB.z, fmaf(x1, wB.w, bl.w)) + s.w + sv.w));

        *(float4*)(out + (size_t)p * OO + o0) = r;
    }
}

extern "C" void kernel_launch(void* const* d_in, const int* in_sizes, int n_in,
                              void* d_out, int out_size, void* d_ws, size_t ws_size,
                              hipStream_t stream) {
    const float* x        = (const float*)d_in[0];
    const float* latent   = (const float*)d_in[1];
    const float* v        = (const float*)d_in[2];
    const float* W_lin    = (const float*)d_in[3];
    const float* b_lin    = (const float*)d_in[4];
    const float* W_shift  = (const float*)d_in[5];
    const float* b_shift  = (const float*)d_in[6];
    const float* W_vshift = (const float*)d_in[7];
    const float* b_vshift = (const float*)d_in[8];
    float* out = (float*)d_out;

    float* smap = (float*)d_ws;                    // [4096, 512] f32 = 8 MB
    float* vmap = smap + (size_t)BB * HWSZ * OO;   // another 8 MB

    // Kernel 1: 32 N-tiles x 256 M-tiles per map, 4 waves/block.
    modmap_wmma_kernel<<<dim3(8, 256, 2), 128, 0, stream>>>(
        latent, v, W_shift, b_shift, W_vshift, b_vshift, smap, vmap);

    // Kernel 2: 262144 points / 32 per block = 8192 blocks.
    siren_gather_kernel<<<dim3(8192), 128, 0, stream>>>(
        x, W_lin, b_lin, smap, vmap, out);
}